// PointcloudGrouping_23974507446931
// MI455X (gfx1250) — compile-verified
//
#include <hip/hip_runtime.h>

#define NPTS   16384
#define NGRP   512
#define GRPSZ  32
#define NBATCH 16

typedef __attribute__((ext_vector_type(2))) float v2f;
typedef __attribute__((ext_vector_type(8))) float v8f;

__device__ __forceinline__ float finf() { return __int_as_float(0x7f800000); }

// ---------------------------------------------------------------------------
// Kernel 1: Farthest Point Sampling. One block per batch; xyz lives packed as
// float4 in CDNA5's 320 KB LDS (256 KB), min_d lives in 16 VGPRs per thread
// (each thread owns points i = tid + t*1024). Inner serial step is
// 1x ds_load_b128 + ~8 VALU per point, no LDS read-modify-write.
// Writes centers [B,512,3] into d_out tail.
// ---------------------------------------------------------------------------
__global__ __launch_bounds__(1024)
void fps_kernel(const float* __restrict__ pts, float* __restrict__ centers)
{
    extern __shared__ float smem[];
    float* sp   = smem;                          // [NPTS][4] = x,y,z,pad
    float* rmax = smem + 4 * NPTS;               // [32]
    int*   ridx = (int*)(smem + 4 * NPTS + 32);  // [32]
    int*   bc   = (int*)(smem + 4 * NPTS + 64);  // [1]

    const int b    = blockIdx.x;
    const int tid  = threadIdx.x;
    const int wave = tid >> 5;
    const int lane = tid & 31;

    #pragma unroll
    for (int t = 0; t < 16; ++t) {
        const int i = tid + (t << 10);
        const float* p = pts + ((size_t)b * NPTS + i) * 6;
        sp[i * 4 + 0] = p[0];
        sp[i * 4 + 1] = p[1];
        sp[i * 4 + 2] = p[2];
        sp[i * 4 + 3] = 0.0f;
    }
    float md[16];
    #pragma unroll
    for (int t = 0; t < 16; ++t) md[t] = finf();
    __syncthreads();

    int last = 0;
    for (int s = 0; s < NGRP; ++s) {
        const float4 c = ((const float4*)sp)[last];  // broadcast b128 load
        if (tid == 0) {
            float* o = centers + ((size_t)b * NGRP + s) * 3;
            o[0] = c.x; o[1] = c.y; o[2] = c.z;
        }
        if (s == NGRP - 1) break;

        // update per-thread min_d (registers) and local argmax
        float bd = -finf();
        int   bi = 0x7fffffff;
        #pragma unroll
        for (int t = 0; t < 16; ++t) {
            const int i = tid + (t << 10);
            const float4 v = ((const float4*)sp)[i];
            const float dx = v.x - c.x, dy = v.y - c.y, dz = v.z - c.z;
            const float d  = dx * dx + dy * dy + dz * dz;
            const float m  = fminf(md[t], d);
            md[t] = m;
            if (m > bd || (m == bd && i < bi)) { bd = m; bi = i; }
        }
        // wave32 shuffle argmax reduce (tie -> lowest index, like jnp.argmax)
        #pragma unroll
        for (int off = 16; off; off >>= 1) {
            const float od = __shfl_down(bd, off);
            const int   oi = __shfl_down(bi, off);
            if (od > bd || (od == bd && oi < bi)) { bd = od; bi = oi; }
        }
        if (lane == 0) { rmax[wave] = bd; ridx[wave] = bi; }
        __syncthreads();
        if (wave == 0) {
            float bd2 = rmax[lane];
            int   bi2 = ridx[lane];
            #pragma unroll
            for (int off = 16; off; off >>= 1) {
                const float od = __shfl_down(bd2, off);
                const int   oi = __shfl_down(bi2, off);
                if (od > bd2 || (od == bd2 && oi < bi2)) { bd2 = od; bi2 = oi; }
            }
            if (lane == 0) *bc = bi2;
        }
        __syncthreads();
        last = *bc;
    }
}

// ---------------------------------------------------------------------------
// Kernel 2: fused distance-matrix (WMMA f32 16x16x4) + streaming top-32
// selection + gather/recenter. One block per (batch, tile of 16 centers).
// 256 threads = 8 waves; compute phase batches 8 global_load_b96 in flight,
// then 8 V_WMMA_F32_16X16X4_F32 (A row = (-2cx,-2cy,-2cz,1),
// B col = (x,y,z,|p|^2)) filling a 16x1024 d2 chunk in LDS; next chunk is
// prefetched (global_prefetch_b8) so L2 latency overlaps the selection phase.
// Selection: one wave per 2 centers, best-32 one-entry-per-lane, threshold +
// ballot stream, final 32-lane bitonic sort by (d, idx) to match top_k order.
// ---------------------------------------------------------------------------
__global__ __launch_bounds__(256)
void knn_group_kernel(const float* __restrict__ pts,
                      const float* __restrict__ centers,
                      float* __restrict__ groups)
{
    extern __shared__ float smem[];
    float* d2buf = smem;             // [16][1024]
    float* cbuf  = smem + 16 * 1024; // [16][3]

    const int b    = blockIdx.y;
    const int gt   = blockIdx.x;     // tile of 16 centers
    const int tid  = threadIdx.x;
    const int wave = tid >> 5;
    const int lane = tid & 31;
    const int half = lane >> 4;
    const int l16  = lane & 15;

    if (tid < 48) cbuf[tid] = centers[((size_t)b * NGRP + gt * 16) * 3 + tid];
    __syncthreads();

    // A-matrix registers (16x4 f32 per ISA layout: vgpr = k&1, lane-half = k>>1)
    float a0, a1;
    {
        const float cx = cbuf[l16 * 3 + 0];
        const float cy = cbuf[l16 * 3 + 1];
        const float cz = cbuf[l16 * 3 + 2];
        a0 = half ? (-2.0f * cz) : (-2.0f * cx); // k=2 : k=0
        a1 = half ? 1.0f         : (-2.0f * cy); // k=3 : k=1
    }
    const v2f va = {a0, a1};

    // per-wave best-32 lists for this wave's 2 centers, one entry per lane
    float bestD[2]; int bestI[2]; float curMax[2];
    bestD[0] = bestD[1] = finf();
    bestI[0] = bestI[1] = 0x7fffffff;
    curMax[0] = curMax[1] = finf();

    for (int chunk = 0; chunk < NPTS / 1024; ++chunk) {
        const int cbase = chunk * 1024;

        // ---- load phase: 8 tiles' coords batched (8 loads in flight) ----
        float px[8], py[8], pz[8];
        #pragma unroll
        for (int t = 0; t < 8; ++t) {
            const int localn = wave * 128 + t * 16 + l16;
            const float* p = pts + ((size_t)b * NPTS + (cbase + localn)) * 6;
            px[t] = p[0]; py[t] = p[1]; pz[t] = p[2];
        }
        // prefetch next chunk's 24 KB of point data into cache
        if (chunk + 1 < NPTS / 1024) {
            const char* nxt =
                (const char*)(pts + ((size_t)b * NPTS + (cbase + 1024)) * 6);
            __builtin_prefetch(nxt + (size_t)tid * 96, 0, 0);
        }

        // ---- compute phase: 8 WMMA tiles -> d2buf[16][1024] ----
        #pragma unroll
        for (int t = 0; t < 8; ++t) {
            const int localn = wave * 128 + t * 16 + l16;
            const float w2 = px[t] * px[t] + py[t] * py[t] + pz[t] * pz[t];
            const float b0 = half ? pz[t] : px[t];  // k=2 : k=0
            const float b1 = half ? w2    : py[t];  // k=3 : k=1
            const v2f vb = {b0, b1};
            v8f acc = {};
            acc = __builtin_amdgcn_wmma_f32_16x16x4_f32(
                false, va, false, vb, (short)0, acc, false, false);
            #pragma unroll
            for (int v = 0; v < 8; ++v) {
                const int row = v + 8 * half;  // C/D layout: vgpr v -> rows v, v+8
                d2buf[row * 1024 + localn] = acc[v];
            }
        }
        __syncthreads();

        // ---- selection phase: wave handles centers 2w, 2w+1 ----
        #pragma unroll
        for (int s = 0; s < 2; ++s) {
            const int cc = wave * 2 + s;
            float cm = curMax[s];
            float ed = bestD[s];
            int   ei = bestI[s];
            for (int i = 0; i < 1024 / 32; ++i) {
                const float dv = d2buf[cc * 1024 + i * 32 + lane];
                const int   gi = cbase + i * 32 + lane;
                unsigned mask = (unsigned)__ballot(dv < cm);
                while (mask) {
                    const int src = __ffs(mask) - 1;
                    mask &= mask - 1;
                    const float cd = __shfl(dv, src);
                    const int   ci = __shfl(gi, src);
                    if (cd < cm) {
                        // evict worst entry (max d, tie -> max idx)
                        float md_ = ed; int mi_ = ei; int ml_ = lane;
                        #pragma unroll
                        for (int off = 16; off; off >>= 1) {
                            const float od = __shfl_xor(md_, off);
                            const int   oi = __shfl_xor(mi_, off);
                            const int   ol = __shfl_xor(ml_, off);
                            if (od > md_ || (od == md_ && oi > mi_)) {
                                md_ = od; mi_ = oi; ml_ = ol;
                            }
                        }
                        if (lane == ml_) { ed = cd; ei = ci; }
                        float mx = ed;
                        #pragma unroll
                        for (int off = 16; off; off >>= 1)
                            mx = fmaxf(mx, __shfl_xor(mx, off));
                        cm = mx;
                    }
                }
            }
            curMax[s] = cm; bestD[s] = ed; bestI[s] = ei;
        }
        __syncthreads();
    }

    // ---- sort each center's 32 entries ascending by (d, idx); gather ----
    #pragma unroll
    for (int s = 0; s < 2; ++s) {
        const int cc = wave * 2 + s;
        float ed = bestD[s];
        int   ei = bestI[s];
        #pragma unroll
        for (int k = 2; k <= 32; k <<= 1) {
            #pragma unroll
            for (int j = k >> 1; j >= 1; j >>= 1) {
                const float od = __shfl_xor(ed, j);
                const int   oi = __shfl_xor(ei, j);
                const bool up        = ((lane & k) == 0);
                const bool lower     = ((lane & j) == 0);
                const bool otherLess = (od < ed) || (od == ed && oi < ei);
                const bool takeOther = (lower == up) ? otherLess : !otherLess;
                if (takeOther) { ed = od; ei = oi; }
            }
        }
        const int g = gt * 16 + cc;
        const float cx = cbuf[cc * 3 + 0];
        const float cy = cbuf[cc * 3 + 1];
        const float cz = cbuf[cc * 3 + 2];
        const float* p = pts + ((size_t)b * NPTS + ei) * 6;
        float* o = groups + (((size_t)b * NGRP + g) * GRPSZ + lane) * 6;
        o[0] = p[0] - cx;
        o[1] = p[1] - cy;
        o[2] = p[2] - cz;
        o[3] = p[3];
        o[4] = p[4];
        o[5] = p[5];
    }
}

// ---------------------------------------------------------------------------
extern "C" void kernel_launch(void* const* d_in, const int* in_sizes, int n_in,
                              void* d_out, int out_size, void* d_ws, size_t ws_size,
                              hipStream_t stream) {
    (void)in_sizes; (void)n_in; (void)out_size; (void)d_ws; (void)ws_size;

    const float* pts = (const float*)d_in[0];
    float* out      = (float*)d_out;
    float* groups   = out;                                      // [16,512,32,6]
    float* centers  = out + (size_t)NBATCH * NGRP * GRPSZ * 6;  // [16,512,3]

    // FPS: one persistent block per batch; 256.3 KB dynamic LDS (fits 320 KB)
    const size_t lds1 = (size_t)(4 * NPTS + 32 + 32 + 8) * sizeof(float);
    fps_kernel<<<NBATCH, 1024, lds1, stream>>>(pts, centers);

    // KNN + grouping: 32 center-tiles x 16 batches, 64.2 KB dynamic LDS
    const size_t lds2 = (size_t)(16 * 1024 + 48) * sizeof(float);
    dim3 grid(NGRP / 16, NBATCH);
    knn_group_kernel<<<grid, 256, lds2, stream>>>(pts, centers, groups);
}